// CIN_layer_71588514890532
// MI455X (gfx1250) — compile-verified
//
#include <hip/hip_runtime.h>

// CDNA5 / gfx1250, wave32. Fused CIN layer:
//   out[b,o,h*32+t] = bias[o] + sum_kh x0[b,t,2kh+h] * ( sum_kw xk[b,kw,2kh+h] * W[o,kh,kw] )
// Step 1 (batched matvec) on VALU, step 2 (per-(b,h) 32x64x32 GEMM) on
// V_WMMA_F32_16X16X4_F32 (exact fp32 — kernel is HBM-bound, no reason to downcast).
// x_0 is staged global->LDS with GLOBAL_LOAD_ASYNC_TO_LDS_B128 (ASYNCcnt),
// overlapped with step 1; waited with s_wait_asynccnt before step 2.

typedef __attribute__((ext_vector_type(2))) float v2f;
typedef __attribute__((ext_vector_type(8))) float v8f;

#define BS    1024   // batches
#define TS    32     // ts0 == tsk
#define FD    64     // features
#define NF    64     // filters
#define NB    4      // batches per workgroup
#define NT    128    // threads per workgroup (4 waves)
#define PAD   34     // xk/Ap row pad (even -> 8B-aligned ds_load_b64)
#define X0PAD 68     // x0 row pad (272B rows -> 16B aligned for async B128)

// LDS layout (floats):
//   s_xk [NB][FD][PAD]      : xkT[b][g][kw]  = x_k[b][kw][g]
//   s_x0 [NB][TS][X0PAD]    : natural x_0[b][t][f] rows (async-B128 friendly)
//   s_ap [NB][2][NF][PAD]   : Ap[b][h][o][kh] = sum_kw xkT[b][2kh+h][kw]*W[o][kh][kw]
//   s_b  [NF]
#define XK_SZ (NB * FD * PAD)
#define X0_SZ (NB * TS * X0PAD)
#define AP_SZ (NB * 2 * NF * PAD)
#define LDS_FLOATS (XK_SZ + X0_SZ + AP_SZ + NF)

#if defined(__has_builtin)
#if __has_builtin(__builtin_amdgcn_global_load_async_to_lds_b128)
#define HAVE_ASYNC_LDS 1
#endif
#endif
#ifndef HAVE_ASYNC_LDS
#define HAVE_ASYNC_LDS 0
#endif

#if HAVE_ASYNC_LDS
// Param types per hipcc diagnostic: 'int __vector(4) __device__ *' (addrspace(1))
// for the global source; LDS side is the addrspace(3) analogue.
typedef int v4i __attribute__((vector_size(4 * sizeof(int))));
typedef __attribute__((address_space(1))) v4i* gv4i_p;
typedef __attribute__((address_space(3))) v4i* lv4i_p;
#endif

__global__ __launch_bounds__(NT)
void cin_fused_kernel(const float* __restrict__ x0,
                      const float* __restrict__ xk,
                      const float* __restrict__ w,
                      const float* __restrict__ bias,
                      float* __restrict__ out)
{
    extern __shared__ float lds[];
    float* s_xk = lds;
    float* s_x0 = lds + XK_SZ;
    float* s_ap = s_x0 + X0_SZ;
    float* s_b  = s_ap + AP_SZ;

    const int tid  = threadIdx.x;
    const int lane = tid & 31;
    const int wv   = tid >> 5;
    const int b0   = blockIdx.x * NB;

    // Warm WGP cache for this wave's first W rows (L2-resident, reused by all WGs).
    __builtin_prefetch(w + ((size_t)wv * TS + lane) * TS, 0, 1);

#if HAVE_ASYNC_LDS
    // ---- kick off x_0 staging on the async global->LDS path (no VGPR traffic,
    //      tracked by ASYNCcnt, overlaps the whole of step 1) ----
    for (int b = 0; b < NB; ++b) {
        const float* gb = x0 + (size_t)(b0 + b) * (TS * FD);
#pragma unroll
        for (int i = 0; i < (TS * FD / 4) / NT; ++i) {   // 4 x B128 per batch per thread
            const int e = (i * NT + tid) * 4;            // flat element, multiple of 4
            const int t = e >> 6, f = e & 63;
            const float* gsrc = gb + e;
            float*       ldst = s_x0 + (b * TS + t) * X0PAD + f;   // 16B aligned
            __builtin_amdgcn_global_load_async_to_lds_b128(
                (gv4i_p)gsrc, (lv4i_p)ldst, /*offset=*/0, /*cpol=*/0);
        }
    }
#endif

    if (tid < NF) s_b[tid] = bias[tid];

    // ---------------- stage x_k transposed (needed immediately by step 1) ---------
    for (int b = 0; b < NB; ++b) {
        const float4* xkb = (const float4*)(xk + (size_t)(b0 + b) * (TS * FD));
#if !HAVE_ASYNC_LDS
        const float4* x0b = (const float4*)(x0 + (size_t)(b0 + b) * (TS * FD));
#endif
#pragma unroll
        for (int i = 0; i < (TS * FD / 4) / NT; ++i) {   // 4 iters
            const int v = i * NT + tid;                   // float4 index
            const int e = v * 4;                          // flat element index
            {   // xk flat = kw*64 + g  ->  s_xk[b][g][kw]
                float4 q = xkb[v];
                const int kw = e >> 6, g = e & 63;
                float* d = s_xk + b * FD * PAD + kw;
                d[(g + 0) * PAD] = q.x;
                d[(g + 1) * PAD] = q.y;
                d[(g + 2) * PAD] = q.z;
                d[(g + 3) * PAD] = q.w;
            }
#if !HAVE_ASYNC_LDS
            {   // fallback: sync x0 staging, natural rows (contiguous b128 ds store)
                float4 q = x0b[v];
                const int t = e >> 6, f = e & 63;
                *(float4*)(s_x0 + (b * TS + t) * X0PAD + f) = q;
            }
#endif
        }
    }
    __syncthreads();

    // ---------------- step 1: A[b][h][o][kh] (VALU dots, W read once per WG) ------
    // wave = fixed o, lanes = consecutive kh -> each wave's W reads cover a
    // contiguous 4KB block; each W row feeds all NB batches x 2 parities.
#pragma unroll 1
    for (int i = 0; i < (NF * TS) / NT; ++i) {           // 16 iters
        const int o  = i * (NT / 32) + wv;
        const int kh = lane;
        if (i + 1 < (NF * TS) / NT)                       // prefetch next W row
            __builtin_prefetch(w + ((size_t)(o + NT / 32) * TS + kh) * TS, 0, 1);
        const float4* wr4 = (const float4*)(w + ((size_t)o * TS + kh) * TS);
        float4 wr[8];
#pragma unroll
        for (int j = 0; j < 8; ++j) wr[j] = wr4[j];
#pragma unroll
        for (int b = 0; b < NB; ++b) {
#pragma unroll
            for (int h = 0; h < 2; ++h) {
                const float* xr = s_xk + b * FD * PAD + (2 * kh + h) * PAD;
                float acc = 0.f;
#pragma unroll
                for (int j = 0; j < 8; ++j) {
                    acc += wr[j].x * xr[4 * j + 0];
                    acc += wr[j].y * xr[4 * j + 1];
                    acc += wr[j].z * xr[4 * j + 2];
                    acc += wr[j].w * xr[4 * j + 3];
                }
                s_ap[((b * 2 + h) * NF + o) * PAD + kh] = acc;   // o-major, conflict-free
            }
        }
    }

#if HAVE_ASYNC_LDS
    // x_0 async copies must have landed in LDS before step 2.
#if defined(__has_builtin) && __has_builtin(__builtin_amdgcn_s_wait_asynccnt)
    __builtin_amdgcn_s_wait_asynccnt(0);
#else
    asm volatile("s_wait_asynccnt 0" ::: "memory");
#endif
#endif
    __syncthreads();

    // ---------------- step 2: out tiles via V_WMMA_F32_16X16X4_F32 ----------------
    // Per (b,h): D(32x64) = X0(32x32 over kh, parity h) @ Ap(32x64).  64 16x16
    // tiles per WG, 16 per wave, 8 K-steps (K=4).  C initialized to bias[o].
    const int m  = lane & 15;            // A-row (t) / B-col (o) within tile
    const int kb = (lane >> 4) << 1;     // K pair base per 16x4 / 4x16 f32 layout
#pragma unroll 1
    for (int j = 0; j < 16; ++j) {
        const int tau = wv * 16 + j;
        const int nt  = tau & 3;
        const int mt  = (tau >> 2) & 1;
        const int bh  = tau >> 3;        // b*2 + h
        const int h   = bh & 1;
        const int b   = bh >> 1;
        const float* x0n = s_x0 + (b * TS) * X0PAD;   // natural [t][f]
        const float* app = s_ap + bh * NF * PAD;      // [o][kh]
        const int t = mt * 16 + m;
        const int o = nt * 16 + m;
        const float bv = s_b[o];
        v8f acc = { bv, bv, bv, bv, bv, bv, bv, bv };
#pragma unroll
        for (int ks = 0; ks < 8; ++ks) {
            const int k0 = ks * 4 + kb;
            // A[m, k0..k0+1] = x0[t, 2k+h] for k = k0, k0+1
            v2f av = { x0n[t * X0PAD + 2 * k0 + h],
                       x0n[t * X0PAD + 2 * k0 + 2 + h] };
            // B[k0..k0+1, n] = Ap[o][k0..k0+1]  (8B aligned -> ds_load_b64)
            v2f bvv = *(const v2f*)(app + o * PAD + k0);
            acc = __builtin_amdgcn_wmma_f32_16x16x4_f32(
                /*neg_a=*/false, av, /*neg_b=*/false, bvv,
                /*c_mod=*/(short)0, acc, /*reuse_a=*/false, /*reuse_b=*/false);
        }
        // D layout: lane holds col n=o, rows M = v + 8*(lane>>4); w' = h*32+mt*16+M
        const int wb = h * 32 + mt * 16 + ((lane >> 4) << 3);
        float* op = out + (size_t)(b0 + b) * (NF * FD) + (size_t)o * FD + wb;
        *(float4*)(op + 0) = make_float4(acc[0], acc[1], acc[2], acc[3]);
        *(float4*)(op + 4) = make_float4(acc[4], acc[5], acc[6], acc[7]);
    }
}

extern "C" void kernel_launch(void* const* d_in, const int* in_sizes, int n_in,
                              void* d_out, int out_size, void* d_ws, size_t ws_size,
                              hipStream_t stream) {
    const float* x0   = (const float*)d_in[0];
    const float* xk   = (const float*)d_in[1];
    const float* w    = (const float*)d_in[2];
    const float* bias = (const float*)d_in[3];
    float* out = (float*)d_out;

    constexpr int lds_bytes = LDS_FLOATS * (int)sizeof(float);   // ~139.5 KB (<320 KB/WG on CDNA5)
    static_assert(lds_bytes < 320 * 1024, "LDS budget");
    // Raise dynamic-LDS cap (host-side attribute, not a stream op -> capture-safe).
    (void)hipFuncSetAttribute((const void*)cin_fused_kernel,
                              hipFuncAttributeMaxDynamicSharedMemorySize, lds_bytes);

    cin_fused_kernel<<<dim3(BS / NB), dim3(NT), lds_bytes, stream>>>(x0, xk, w, bias, out);
}